// social_stgcn_46462956208716
// MI455X (gfx1250) — compile-verified
//
#include <hip/hip_runtime.h>

typedef __attribute__((ext_vector_type(2))) float v2f;
typedef __attribute__((ext_vector_type(8))) float v8f;

#define NPAD   16384   // padded node count
#define KDIM   16384   // input feature dim
#define XROWS  12288   // real rows of x (rest are zero padding)
#define NEDGE  524288  // number of edges

// ---------- degree / normalization ----------
__global__ void deg_kernel(const int* __restrict__ dst, unsigned* __restrict__ deg) {
  int e = blockIdx.x * blockDim.x + threadIdx.x;
  if (e < NEDGE) atomicAdd(&deg[dst[e]], 1u);
}

__global__ void dinv_kernel(const unsigned* __restrict__ deg, float* __restrict__ dinv) {
  int n = blockIdx.x * blockDim.x + threadIdx.x;
  if (n < NPAD) dinv[n] = rsqrtf((float)(deg[n] + 1u)); // +1 self-loop; deg>=1 always
}

// ---------- pack W1 [16384,5] into WMMA B-operand layout, zero-padded to 16 cols ----------
// Bpack[kb*64 + lane*2 + j] = B[k_local][n], k_local = (lane>>4)*2 + j, n = lane&15
__global__ void bpack_kernel(const float* __restrict__ W1, float* __restrict__ Bpack) {
  int idx = blockIdx.x * blockDim.x + threadIdx.x;
  if (idx >= KDIM * 16) return;
  int kb   = idx >> 6;
  int r    = idx & 63;
  int lane = r >> 1;
  int j    = r & 1;
  int k    = kb * 4 + ((lane >> 4) * 2 + j);
  int n    = lane & 15;
  Bpack[idx] = (n < 5) ? W1[k * 5 + n] : 0.0f;
}

// ---------- h = x_pad @ W1pad  via V_WMMA_F32_16X16X4_F32 ----------
// one wave -> one 16x16 output tile; K loop of 4096 steps (2 b64 loads + 1 wmma each)
__global__ __launch_bounds__(256) void gemm1_wmma(const float* __restrict__ x,
                                                  const float* __restrict__ Bpack,
                                                  float* __restrict__ h) {
  int wave    = (blockIdx.x * blockDim.x + threadIdx.x) >> 5;
  int lane    = threadIdx.x & 31;
  int rowBase = wave * 16;

  v8f c0 = {};
  v8f c1 = {};
  if (rowBase < XROWS) {
    int m  = lane & 15;
    int kh = (lane >> 4) * 2;            // K offset within 4-wide step: 0 or 2
    const float* aptr = x + (size_t)(rowBase + m) * KDIM + kh;
    const float* bptr = Bpack + lane * 2;
    #pragma unroll 4
    for (int kb = 0; kb < KDIM / 4; kb += 2) {
      v2f a0 = *(const v2f*)(aptr);
      v2f b0 = *(const v2f*)(bptr);
      v2f a1 = *(const v2f*)(aptr + 4);
      v2f b1 = *(const v2f*)(bptr + 64);
      c0 = __builtin_amdgcn_wmma_f32_16x16x4_f32(false, a0, false, b0, (short)0, c0, false, false);
      c1 = __builtin_amdgcn_wmma_f32_16x16x4_f32(false, a1, false, b1, (short)0, c1, false, false);
      aptr += 8;
      bptr += 128;
    }
  }
  // store D (16x16 f32 layout: VGPR r -> M = (lane>>4)*8 + r, N = lane&15)
  int n     = lane & 15;
  int mBase = (lane >> 4) * 8;
  float* hp = h + (size_t)rowBase * 16 + n;
  #pragma unroll
  for (int r = 0; r < 8; ++r) {
    hp[(size_t)(mBase + r) * 16] = c0[r] + c1[r];
  }
}

// ---------- hs[n][c] = dinv[n] * h[n][c]  (source-side normalization) ----------
__global__ void scale1_kernel(const float* __restrict__ h, const float* __restrict__ dinv,
                              float* __restrict__ hs) {
  int n = blockIdx.x * blockDim.x + threadIdx.x;
  if (n >= NPAD) return;
  float d = dinv[n];
  #pragma unroll
  for (int c = 0; c < 5; ++c) hs[n * 8 + c] = d * h[n * 16 + c];
}

// ---------- edge scatter-add, layer 1 ----------
__global__ void agg1_kernel(const int* __restrict__ src, const int* __restrict__ dst,
                            const float* __restrict__ hs, float* __restrict__ agg) {
  int e = blockIdx.x * blockDim.x + threadIdx.x;
  if (e >= NEDGE) return;
  int s = src[e], d = dst[e];
  #pragma unroll
  for (int c = 0; c < 5; ++c) unsafeAtomicAdd(&agg[d * 8 + c], hs[s * 8 + c]);
}

// ---------- combine L1 (dst norm + self loop + bias + relu) fused with L2 linear + src norm ----------
__global__ void combine1_kernel(const float* __restrict__ agg, const float* __restrict__ hs,
                                const float* __restrict__ dinv, const float* __restrict__ b1,
                                const float* __restrict__ W2, float* __restrict__ hs2) {
  int n = blockIdx.x * blockDim.x + threadIdx.x;
  if (n >= NPAD) return;
  float d   = dinv[n];
  float acc = 0.0f;
  #pragma unroll
  for (int c = 0; c < 5; ++c) {
    float v = d * (agg[n * 8 + c] + hs[n * 8 + c]) + b1[c]; // self-loop msg = dinv^2*h
    v = fmaxf(v, 0.0f);                                     // relu
    acc += v * W2[c];                                       // layer-2 linear [5 -> 1]
  }
  hs2[n] = d * acc;                                         // pre-scale for layer-2 scatter
}

// ---------- edge scatter-add, layer 2 ----------
__global__ void agg2_kernel(const int* __restrict__ src, const int* __restrict__ dst,
                            const float* __restrict__ hs2, float* __restrict__ agg2) {
  int e = blockIdx.x * blockDim.x + threadIdx.x;
  if (e >= NEDGE) return;
  unsafeAtomicAdd(&agg2[dst[e]], hs2[src[e]]);
}

// ---------- final: dst norm + self loop + bias, then width-1 log_softmax (v - lse(v) = v - v) ----------
__global__ void final_kernel(const float* __restrict__ agg2, const float* __restrict__ hs2,
                             const float* __restrict__ dinv, const float* __restrict__ b2,
                             float* __restrict__ out) {
  int n = blockIdx.x * blockDim.x + threadIdx.x;
  if (n >= NPAD) return;
  float v = dinv[n] * (agg2[n] + hs2[n]) + b2[0];
  // log_softmax over a single-element row: logsumexp(v) == v exactly
  out[n] = v - v;
}

extern "C" void kernel_launch(void* const* d_in, const int* in_sizes, int n_in,
                              void* d_out, int out_size, void* d_ws, size_t ws_size,
                              hipStream_t stream) {
  const float* x   = (const float*)d_in[0];
  const int*   ei  = (const int*)d_in[1];
  const float* W1  = (const float*)d_in[2];
  const float* b1  = (const float*)d_in[3];
  const float* W2  = (const float*)d_in[4];
  const float* b2  = (const float*)d_in[5];
  float*       out = (float*)d_out;

  const int* src = ei;           // edge_index[0]
  const int* dst = ei + NEDGE;   // edge_index[1]

  // workspace layout (~3.3 MB)
  char* ws = (char*)d_ws;
  unsigned* deg  = (unsigned*)ws;  ws += (size_t)NPAD * 4;
  float* dinv    = (float*)ws;     ws += (size_t)NPAD * 4;
  float* Bpack   = (float*)ws;     ws += (size_t)KDIM * 16 * 4;
  float* h       = (float*)ws;     ws += (size_t)NPAD * 16 * 4;
  float* hs      = (float*)ws;     ws += (size_t)NPAD * 8 * 4;
  float* agg     = (float*)ws;     ws += (size_t)NPAD * 8 * 4;
  float* hs2     = (float*)ws;     ws += (size_t)NPAD * 4;
  float* agg2    = (float*)ws;     ws += (size_t)NPAD * 4;

  // accumulators must be zeroed every call (graph-capture-safe async memsets)
  hipMemsetAsync(deg,  0, (size_t)NPAD * 4,     stream);
  hipMemsetAsync(agg,  0, (size_t)NPAD * 8 * 4, stream);
  hipMemsetAsync(agg2, 0, (size_t)NPAD * 4,     stream);

  deg_kernel    <<<NEDGE / 256, 256, 0, stream>>>(dst, deg);
  dinv_kernel   <<<NPAD / 256,  256, 0, stream>>>(deg, dinv);
  bpack_kernel  <<<(KDIM * 16) / 256, 256, 0, stream>>>(W1, Bpack);
  gemm1_wmma    <<<(NPAD / 16) / 8, 256, 0, stream>>>(x, Bpack, h);
  scale1_kernel <<<NPAD / 256,  256, 0, stream>>>(h, dinv, hs);
  agg1_kernel   <<<NEDGE / 256, 256, 0, stream>>>(src, dst, hs, agg);
  combine1_kernel<<<NPAD / 256, 256, 0, stream>>>(agg, hs, dinv, b1, W2, hs2);
  agg2_kernel   <<<NEDGE / 256, 256, 0, stream>>>(src, dst, hs2, agg2);
  final_kernel  <<<NPAD / 256,  256, 0, stream>>>(agg2, hs2, dinv, b2, out);
}